// PositionalEncoder_6451040878909
// MI455X (gfx1250) — compile-verified
//
#include <hip/hip_runtime.h>
#include <math.h>

// out[b,t,d] = x[b,t,d] + pe[t,d]
//   pe[t,d] = (d even) ? sin(a) : cos(a),  a = t / 10000^(2d/D)
//
// Memory-bound streaming op: 256 MB in + 256 MB out => ~22 us floor @ 23.3 TB/s.
// PE recomputed in-register once per thread, amortized 32x over the batch.
// Input stream goes through the gfx1250 async global->LDS engine
// (GLOBAL_LOAD_ASYNC_TO_LDS_B128 / ASYNCcnt) in a 4-deep pipeline; each lane
// consumes only the LDS bytes it fetched, so no barriers, only s_wait_asynccnt.
// Output uses non-temporal b128 stores (zero reuse -> keep L2 clean).

typedef __attribute__((ext_vector_type(4))) float v4f;

#define PE_D_MODEL 1024
#define PE_TSTEPS  2048
#define PE_BATCH   32
#define PIPE_DEPTH 4   // must be power of two

__device__ __forceinline__ void async_tile_load(unsigned ldsByteAddr, const float* gsrc) {
    // VDST = per-lane LDS byte address, VADDR = per-lane 64-bit global address (GV mode).
    asm volatile("global_load_async_to_lds_b128 %0, %1, off"
                 :
                 : "v"(ldsByteAddr), "v"(gsrc)
                 : "memory");
}

__device__ __forceinline__ void wait_async_le(int n) {
#if __has_builtin(__builtin_amdgcn_s_wait_asynccnt)
    switch (n) {
    case 0:  __builtin_amdgcn_s_wait_asynccnt(0); break;
    case 1:  __builtin_amdgcn_s_wait_asynccnt(1); break;
    case 2:  __builtin_amdgcn_s_wait_asynccnt(2); break;
    default: __builtin_amdgcn_s_wait_asynccnt(3); break;
    }
#else
    switch (n) {
    case 0:  asm volatile("s_wait_asynccnt 0x0" ::: "memory"); break;
    case 1:  asm volatile("s_wait_asynccnt 0x1" ::: "memory"); break;
    case 2:  asm volatile("s_wait_asynccnt 0x2" ::: "memory"); break;
    default: asm volatile("s_wait_asynccnt 0x3" ::: "memory"); break;
    }
#endif
}

__global__ __launch_bounds__(256) void pe_add_async_kernel(const float* __restrict__ x,
                                                           float* __restrict__ out) {
    __shared__ v4f tile[PIPE_DEPTH][256];   // 4 stages x 4 KB = 16 KB LDS

    const int t  = blockIdx.x;          // 0..2047 : one timestep row per block
    const int d0 = threadIdx.x << 2;    // 0..1020 : 4 consecutive channels per thread

    // angle(t,d) = t * 10000^(-2d/D) = t * exp2(-d * 2*log2(10000)/D)
    const float kC = -0.025952563241311463f;
    const float tf = (float)t;

    // d0 is a multiple of 4 -> components 0,2 even (sin), 1,3 odd (cos)
    v4f pe;
    pe.x = sinf(tf * exp2f(kC * (float)(d0 + 0)));
    pe.y = cosf(tf * exp2f(kC * (float)(d0 + 1)));
    pe.z = sinf(tf * exp2f(kC * (float)(d0 + 2)));
    pe.w = cosf(tf * exp2f(kC * (float)(d0 + 3)));

    const size_t rowBase = (size_t)t * PE_D_MODEL + (size_t)d0;   // in floats
    const size_t bStride = (size_t)PE_TSTEPS * PE_D_MODEL;        // 8 MB in floats

    // Low 32 bits of the generic shared-memory address are the LDS byte
    // address (flat->LDS aperture mapping discards the upper bits).
    const unsigned myLds = (unsigned)(size_t)(const void*)&tile[0][threadIdx.x];
    const unsigned stageBytes = (unsigned)sizeof(tile[0]);        // 4096

    // ---- prologue: prime PIPE_DEPTH-1 stages ----
#pragma unroll
    for (int b = 0; b < PIPE_DEPTH - 1; ++b) {
        async_tile_load(myLds + (unsigned)b * stageBytes,
                        x + rowBase + (size_t)b * bStride);
    }

    // ---- steady state: issue b+3, wait <=3, consume b ----
#pragma unroll
    for (int b = 0; b < PE_BATCH; ++b) {
        const int bn = b + (PIPE_DEPTH - 1);
        if (bn < PE_BATCH) {
            async_tile_load(myLds + (unsigned)(bn & (PIPE_DEPTH - 1)) * stageBytes,
                            x + rowBase + (size_t)bn * bStride);
            wait_async_le(PIPE_DEPTH - 1);        // oldest stage (b) complete
        } else {
            wait_async_le(PE_BATCH - 1 - b);      // drain: 2, 1, 0
        }

        v4f v = tile[b & (PIPE_DEPTH - 1)][threadIdx.x];   // ds_load_b128
        v += pe;
        __builtin_nontemporal_store(v, (v4f*)(out + rowBase + (size_t)b * bStride));
    }
}

extern "C" void kernel_launch(void* const* d_in, const int* in_sizes, int n_in,
                              void* d_out, int out_size, void* d_ws, size_t ws_size,
                              hipStream_t stream) {
    (void)in_sizes; (void)n_in; (void)out_size; (void)d_ws; (void)ws_size;

    const float* x   = (const float*)d_in[0];
    float*       out = (float*)d_out;

    // 2048 blocks (one per timestep) x 256 threads (4 floats each -> D=1024).
    dim3 grid(PE_TSTEPS);
    dim3 block(PE_D_MODEL / 4);
    pe_add_async_kernel<<<grid, block, 0, stream>>>(x, out);
}